// TRTModel_post_6193342841265
// MI455X (gfx1250) — compile-verified
//
#include <hip/hip_runtime.h>
#include <math.h>

#define NUM_CLASSES 10
#define BOX_CODE 7
#define NUM_ANCH 4
#define HWSZ 250000          // 500*500
#define NTOT 1000000         // HWSZ * NUM_ANCH
#define CAND_CAP 4096

#if defined(__has_builtin)
#if __has_builtin(__builtin_amdgcn_global_load_async_to_lds_b128) && \
    __has_builtin(__builtin_amdgcn_s_wait_asynccnt)
#define USE_ASYNC_LDS 1
#endif
#endif

#ifdef USE_ASYNC_LDS
typedef int b128_t __attribute__((vector_size(16)));
typedef __attribute__((address_space(1))) b128_t* gb128_p;
typedef __attribute__((address_space(3))) b128_t* lb128_p;
#endif

__device__ __forceinline__ unsigned key_of(float f) {
  unsigned u = __float_as_uint(f);
  return (u & 0x80000000u) ? ~u : (u | 0x80000000u);
}

__global__ void zero_kernel(unsigned* __restrict__ p, int n) {
  int i = blockIdx.x * blockDim.x + threadIdx.x;
  if (i < n) p[i] = 0u;
}

// One thread per (h,w) location; computes 4 anchor keys (max over 10 classes,
// monotone u32 transform) and builds the high-16-bit histogram.
// cls tile (40 channels x 256 locations) is staged into LDS with CDNA5 async
// global->LDS copies when the builtins are available.
__global__ __launch_bounds__(256) void keys_hist_kernel(
    const float* __restrict__ cls, unsigned* __restrict__ keys,
    unsigned* __restrict__ hist) {
  __shared__ float tile[NUM_ANCH * NUM_CLASSES * 256];
  const int t = threadIdx.x;
  const long long blockStart = (long long)blockIdx.x * 256;

#ifdef USE_ASYNC_LDS
  // 10 async b128 issues per lane cover the 40x256 float tile (channel-major).
  #pragma unroll
  for (int i = 0; i < 10; ++i) {
    int e4 = i * 1024 + t * 4;      // flat float index within tile, mult of 4
    int ch = e4 >> 8;               // 0..39
    int col = e4 & 255;             // multiple of 4
    long long g = blockStart + col;
    if (g > (long long)HWSZ - 4) g = (long long)HWSZ - 4;  // clamp tail block
    const float* gp = cls + (long long)ch * HWSZ + g;
    __builtin_amdgcn_global_load_async_to_lds_b128(
        (gb128_p)(gp), (lb128_p)(&tile[ch * 256 + col]), 0, 0);
  }
  __builtin_amdgcn_s_wait_asynccnt(0);
  __syncthreads();
#else
  for (int ch = 0; ch < NUM_ANCH * NUM_CLASSES; ++ch) {
    long long g = blockStart + t;
    if (g >= HWSZ) g = HWSZ - 1;
    tile[ch * 256 + t] = cls[(long long)ch * HWSZ + g];
  }
  __syncthreads();
#endif

  long long loc = blockStart + t;
  if (loc < HWSZ) {
    unsigned kk[NUM_ANCH];
    #pragma unroll
    for (int a = 0; a < NUM_ANCH; ++a) {
      float m = tile[(a * NUM_CLASSES) * 256 + t];
      #pragma unroll
      for (int c = 1; c < NUM_CLASSES; ++c)
        m = fmaxf(m, tile[(a * NUM_CLASSES + c) * 256 + t]);
      kk[a] = key_of(m);
      atomicAdd(&hist[kk[a] >> 16], 1u);
    }
    ((uint4*)keys)[loc] = make_uint4(kk[0], kk[1], kk[2], kk[3]);
  }
}

// Single block, 1024 threads (32 wave32 waves). Suffix-scan 65536 bins
// (64 per thread) from the top to find the bin containing the K-th largest.
// pass 0: high 16 bits -> ctrl[0]=bin, ctrl[1]=count strictly above; zero hist.
// pass 1: low 16 bits  -> ctrl[2]=exact 32-bit threshold T.
__global__ __launch_bounds__(1024) void scan_kernel(unsigned* __restrict__ hist,
                                                    unsigned* __restrict__ ctrl,
                                                    int K, int pass) {
  __shared__ unsigned chunkSum[1024];
  __shared__ unsigned scanBuf[1024];
  const int t = threadIdx.x;
  const unsigned base = (unsigned)t * 64u;
  const unsigned Keff = (pass == 0) ? (unsigned)K : ((unsigned)K - ctrl[1]);

  unsigned s = 0;
  for (int i = 0; i < 64; ++i) s += hist[base + i];
  chunkSum[t] = s;
  unsigned v = s;
  scanBuf[t] = v;
  __syncthreads();
  for (int off = 1; off < 1024; off <<= 1) {   // inclusive suffix scan
    unsigned add = (t + off < 1024) ? scanBuf[t + off] : 0u;
    __syncthreads();
    v += add;
    scanBuf[t] = v;
    __syncthreads();
  }
  unsigned sufIncl = v;
  unsigned sufExcl = v - chunkSum[t];
  if (sufExcl < Keff && sufIncl >= Keff) {     // exactly one thread
    unsigned run = sufExcl;
    int bsel = 0;
    for (int b = 63; b >= 0; --b) {
      unsigned h = hist[base + b];
      if (run + h >= Keff) { bsel = b; break; }
      run += h;
    }
    unsigned binIdx = base + (unsigned)bsel;
    if (pass == 0) { ctrl[0] = binIdx; ctrl[1] = run; }
    else           { ctrl[2] = (ctrl[0] << 16) | binIdx; }
  }
  __syncthreads();
  if (pass == 0) {                             // clear hist for pass 1
    for (int i = 0; i < 64; ++i) hist[base + i] = 0u;
  }
}

__global__ void hist2_kernel(const unsigned* __restrict__ keys,
                             unsigned* __restrict__ hist,
                             const unsigned* __restrict__ ctrl) {
  int i = blockIdx.x * blockDim.x + threadIdx.x;
  if (i >= NTOT) return;
  unsigned hb = ctrl[0];
  unsigned k = keys[i];
  if ((k >> 16) == hb) atomicAdd(&hist[k & 0xFFFFu], 1u);
}

__global__ void compact_kernel(const unsigned* __restrict__ keys,
                               unsigned* __restrict__ ctrl,
                               unsigned long long* __restrict__ cand) {
  int i = blockIdx.x * blockDim.x + threadIdx.x;
  if (i >= NTOT) return;
  __builtin_prefetch(keys + i + 16384, 0, 1);
  unsigned k = keys[i];
  unsigned T = ctrl[2];
  if (k >= T) {
    unsigned pos = atomicAdd(&ctrl[4], 1u);
    if (pos < CAND_CAP)
      cand[pos] = ((unsigned long long)k << 32) | (unsigned)(~(unsigned)i);
  }
}

// Bitonic sort of up to 4096 composite keys (desc score, asc index) in LDS.
__global__ __launch_bounds__(1024) void sort_kernel(
    const unsigned long long* __restrict__ cand,
    const unsigned* __restrict__ ctrl, unsigned* __restrict__ sortedIdx) {
  __shared__ unsigned long long sm[CAND_CAP];
  const int t = threadIdx.x;
  unsigned n = ctrl[4];
  if (n > CAND_CAP) n = CAND_CAP;
  for (int i = t; i < CAND_CAP; i += 1024) sm[i] = (i < (int)n) ? cand[i] : 0ull;
  __syncthreads();
  for (unsigned k = 2; k <= CAND_CAP; k <<= 1) {
    for (unsigned j = k >> 1; j > 0; j >>= 1) {
      #pragma unroll
      for (unsigned ss = 0; ss < CAND_CAP / 1024; ++ss) {
        unsigned i = (unsigned)t + ss * 1024u;
        unsigned ixj = i ^ j;
        if (ixj > i) {
          unsigned long long a = sm[i], b = sm[ixj];
          bool descSeg = ((i & k) == 0u);
          if ((a < b) == descSeg) { sm[i] = b; sm[ixj] = a; }
        }
      }
      __syncthreads();
    }
  }
  sortedIdx[t] = ~(unsigned)(sm[t] & 0xFFFFFFFFull);
}

__global__ void decode_kernel(const float* __restrict__ cls,
                              const float* __restrict__ bbox,
                              const float* __restrict__ dirp,
                              const float* __restrict__ anchors,
                              const unsigned* __restrict__ sortedIdx,
                              float* __restrict__ out, int K) {
  int r = blockIdx.x * blockDim.x + threadIdx.x;
  if (r >= K) return;
  unsigned n = sortedIdx[r];
  if (n >= (unsigned)NTOT) n = 0u;
  unsigned loc = n >> 2;
  unsigned a = n & 3u;

  #pragma unroll
  for (int c = 0; c < NUM_CLASSES; ++c) {
    float x = cls[(size_t)(a * NUM_CLASSES + c) * HWSZ + loc];
    out[(size_t)r * NUM_CLASSES + c] = 1.0f / (1.0f + expf(-x));
  }

  float d[BOX_CODE];
  #pragma unroll
  for (int j = 0; j < BOX_CODE; ++j)
    d[j] = bbox[(size_t)(a * BOX_CODE + j) * HWSZ + loc];

  const float* an = anchors + (size_t)n * BOX_CODE;
  float xa = an[0], ya = an[1], za = an[2], wa = an[3], la = an[4],
        ha = an[5], ra = an[6];
  za += 0.5f * ha;
  float diag = sqrtf(la * la + wa * wa);
  float xg = fmaf(d[0], diag, xa);
  float yg = fmaf(d[1], diag, ya);
  float zg = fmaf(d[2], ha, za);
  float wg = expf(d[3]) * wa;
  float lg = expf(d[4]) * la;
  float hg = expf(d[5]) * ha;
  float rg = d[6] + ra;
  zg -= 0.5f * hg;

  float* ob = out + (size_t)K * NUM_CLASSES + (size_t)r * BOX_CODE;
  ob[0] = xg; ob[1] = yg; ob[2] = zg; ob[3] = wg; ob[4] = lg; ob[5] = hg; ob[6] = rg;

  float d0 = dirp[(size_t)(a * 2 + 0) * HWSZ + loc];
  float d1 = dirp[(size_t)(a * 2 + 1) * HWSZ + loc];
  out[(size_t)K * (NUM_CLASSES + BOX_CODE) + r] = (d1 > d0) ? 1.0f : 0.0f;
}

extern "C" void kernel_launch(void* const* d_in, const int* in_sizes, int n_in,
                              void* d_out, int out_size, void* d_ws,
                              size_t ws_size, hipStream_t stream) {
  (void)in_sizes; (void)n_in; (void)ws_size;
  const float* cls = (const float*)d_in[0];
  const float* bbox = (const float*)d_in[1];
  const float* dirp = (const float*)d_in[2];
  const float* anchors = (const float*)d_in[3];
  int K = out_size / (NUM_CLASSES + BOX_CODE + 1);  // 18000/18 = 1000
  if (K > 1024) K = 1024;
  if (K < 1) K = 1;

  // workspace layout (u32 units): keys[1M] | hist[65536] | ctrl[16] |
  // sortedIdx[1024] | cand[4096 u64]
  unsigned* ws = (unsigned*)d_ws;
  unsigned* keys = ws;
  unsigned* hist = ws + NTOT;
  unsigned* ctrl = hist + 65536;
  unsigned* sortedIdx = ctrl + 16;
  unsigned long long* cand = (unsigned long long*)(sortedIdx + 1024);
  float* out = (float*)d_out;

  zero_kernel<<<(65536 + 16 + 255) / 256, 256, 0, stream>>>(hist, 65536 + 16);
  keys_hist_kernel<<<(HWSZ + 255) / 256, 256, 0, stream>>>(cls, keys, hist);
  scan_kernel<<<1, 1024, 0, stream>>>(hist, ctrl, K, 0);
  hist2_kernel<<<(NTOT + 255) / 256, 256, 0, stream>>>(keys, hist, ctrl);
  scan_kernel<<<1, 1024, 0, stream>>>(hist, ctrl, K, 1);
  compact_kernel<<<(NTOT + 255) / 256, 256, 0, stream>>>(keys, ctrl, cand);
  sort_kernel<<<1, 1024, 0, stream>>>(cand, ctrl, sortedIdx);
  decode_kernel<<<(K + 255) / 256, 256, 0, stream>>>(cls, bbox, dirp, anchors,
                                                     sortedIdx, out, K);
}